// SSIMLoss_28690381537948
// MI455X (gfx1250) — compile-verified
//
#include <hip/hip_runtime.h>

// SSIM loss, fused single-pass per tile, f32 WMMA (V_WMMA_F32_16X16X4_F32)
// B=64, C=1, H=W=512. Tile = 64x64 outputs per 256-thread block (8 wave32).

typedef __attribute__((ext_vector_type(2))) float v2f;
typedef __attribute__((ext_vector_type(8))) float v8f;

#define IMG_DIM   512
#define TILE      64
#define HALO      5
#define REGION    74            // TILE + 2*HALO
#define IN_ROWS   80            // padded rows (5 blocks of 16)
#define IN_STRIDE 76            // padded row stride (floats)
#define H_ROWS    76            // H rows kept (only rows 0..75 are read)
#define H_STRIDE  72            // stride chosen so k and k+2 rows hit disjoint banks

__device__ __forceinline__ v8f wmma4(v2f a, v2f b, v8f c) {
  // D = A(16x4,f32) * B(4x16,f32) + C(16x16,f32)
  return __builtin_amdgcn_wmma_f32_16x16x4_f32(
      /*neg_a=*/false, a, /*neg_b=*/false, b,
      /*c_mod=*/(short)0, c, /*reuse_a=*/false, /*reuse_b=*/false);
}

__global__ __launch_bounds__(256) void ssim_main_kernel(
    const float* __restrict__ img1, const float* __restrict__ img2,
    float* __restrict__ partials) {
  __shared__ float sIn[2][IN_ROWS * IN_STRIDE];   // 48.6 KB
  __shared__ float sH[5][H_ROWS * H_STRIDE];      // 109.4 KB
  __shared__ float sRed[256];

  const int tid  = threadIdx.x;
  const int wave = tid >> 5;
  const int lane = tid & 31;
  const int half = lane >> 4;   // 0: lanes 0-15, 1: lanes 16-31
  const int l16  = lane & 15;

  const int bimg   = blockIdx.x >> 6;   // 64 images
  const int tileId = blockIdx.x & 63;   // 8x8 tiles per image
  const int r0 = (tileId >> 3) << 6;
  const int c0 = (tileId & 7) << 6;

  const float* base1 = img1 + (size_t)bimg * (IMG_DIM * IMG_DIM);
  const float* base2 = img2 + (size_t)bimg * (IMG_DIM * IMG_DIM);

  // ---------- load tile + halo into LDS; zero outside image / pad ----------
  for (int i = tid; i < IN_ROWS * IN_STRIDE; i += 256) {
    int row = i / IN_STRIDE, col = i % IN_STRIDE;
    int gr = r0 - HALO + row, gc = c0 - HALO + col;
    float v1 = 0.0f, v2 = 0.0f;
    if (row < REGION && col < REGION && gr >= 0 && gr < IMG_DIM && gc >= 0 &&
        gc < IMG_DIM) {
      size_t off = (size_t)gr * IMG_DIM + gc;
      v1 = base1[off];
      v2 = base2[off];
    }
    sIn[0][i] = v1;
    sIn[1][i] = v2;
  }

  // ---------- Gaussian weights (sigma=1.5, 11 taps, normalized) ----------
  float g[11];
  {
    float s = 0.0f;
    #pragma unroll
    for (int j = 0; j < 11; ++j) {
      float d = (float)(j - 5);
      g[j] = __expf(-(d * d) / 4.5f);
      s += g[j];
    }
    float inv = 1.0f / s;
    #pragma unroll
    for (int j = 0; j < 11; ++j) g[j] *= inv;
  }

  // Constant band-matrix fragments. For K-chunk t, lane holds value g[k - l16]
  // with k = t*4 + 2*half + i. Identical register image serves as:
  //   B (4x16) fragment for the horizontal pass (B[k][n] = g[k-n]), and
  //   A (16x4) fragment for the vertical pass   (A[m][k] = g[k-m]).
  v2f W[7];
  #pragma unroll
  for (int t = 0; t < 7; ++t) {
    #pragma unroll
    for (int i = 0; i < 2; ++i) {
      int k = t * 4 + half * 2 + i;
      int d = k - l16;
      W[t][i] = (k < 26 && d >= 0 && d <= 10) ? g[d] : 0.0f;
    }
  }

  __syncthreads();

  // ---------- horizontal pass: 5 row-blocks x 4 col-blocks = 20 tasks ------
  for (int task = wave; task < 20; task += 8) {
    int rb = task >> 2;   // 0..4  (rows rb*16 .. rb*16+15 of padded region)
    int cb = task & 3;    // 0..3  (output cols cb*16 .. cb*16+15)
    v8f acc[5];
    #pragma unroll
    for (int q = 0; q < 5; ++q)
      #pragma unroll
      for (int v = 0; v < 8; ++v) acc[q][v] = 0.0f;

    #pragma unroll
    for (int t = 0; t < 7; ++t) {
      int rowA = rb * 16 + l16;
      int colA = cb * 16 + t * 4 + half * 2;   // input-region column (K index)
      const float* p1 = &sIn[0][rowA * IN_STRIDE + colA];
      const float* p2 = &sIn[1][rowA * IN_STRIDE + colA];
      v2f a1, a2;
      a1[0] = p1[0]; a1[1] = p1[1];
      a2[0] = p2[0]; a2[1] = p2[1];
      v2f a11 = a1 * a1, a22 = a2 * a2, a12 = a1 * a2;
      acc[0] = wmma4(a1,  W[t], acc[0]);
      acc[1] = wmma4(a2,  W[t], acc[1]);
      acc[2] = wmma4(a11, W[t], acc[2]);
      acc[3] = wmma4(a22, W[t], acc[3]);
      acc[4] = wmma4(a12, W[t], acc[4]);
    }
    // D layout: VGPR v, lanes 0-15 -> M=v ; lanes 16-31 -> M=v+8
    #pragma unroll
    for (int q = 0; q < 5; ++q)
      #pragma unroll
      for (int v = 0; v < 8; ++v) {
        int row = rb * 16 + v + half * 8;
        if (row < H_ROWS) sH[q][row * H_STRIDE + cb * 16 + l16] = acc[q][v];
      }
  }

  __syncthreads();

  // ---------- vertical pass + SSIM map: 4x4 = 16 tasks, 2 per wave ---------
  float psum = 0.0f;
  for (int task = wave * 2; task < wave * 2 + 2; ++task) {
    int rb = task >> 2, cb = task & 3;
    v8f acc[5];
    #pragma unroll
    for (int q = 0; q < 5; ++q)
      #pragma unroll
      for (int v = 0; v < 8; ++v) acc[q][v] = 0.0f;

    #pragma unroll
    for (int t = 0; t < 7; ++t) {
      int kb = rb * 16 + t * 4 + half * 2;   // H-region row (K index)
      #pragma unroll
      for (int q = 0; q < 5; ++q) {
        v2f bf;
        bf[0] = sH[q][(kb + 0) * H_STRIDE + cb * 16 + l16];
        bf[1] = sH[q][(kb + 1) * H_STRIDE + cb * 16 + l16];
        acc[q] = wmma4(W[t], bf, acc[q]);
      }
    }

    const float C1 = 1e-4f, C2 = 9e-4f;
    #pragma unroll
    for (int v = 0; v < 8; ++v) {
      float mu1 = acc[0][v], mu2 = acc[1][v];
      float e11 = acc[2][v], e22 = acc[3][v], e12 = acc[4][v];
      float mu1s = mu1 * mu1, mu2s = mu2 * mu2, m12 = mu1 * mu2;
      float s1 = e11 - mu1s, s2 = e22 - mu2s, s12 = e12 - m12;
      float num = (2.0f * m12 + C1) * (2.0f * s12 + C2);
      float den = (mu1s + mu2s + C1) * (s1 + s2 + C2);
      psum += num / den;
    }
  }

  // ---------- block reduction -> one partial per block ----------
  sRed[tid] = psum;
  __syncthreads();
  #pragma unroll
  for (int s = 128; s > 0; s >>= 1) {
    if (tid < s) sRed[tid] += sRed[tid + s];
    __syncthreads();
  }
  if (tid == 0) partials[blockIdx.x] = sRed[0];
}

__global__ __launch_bounds__(256) void ssim_finalize_kernel(
    const float* __restrict__ partials, float* __restrict__ out, int nblocks) {
  __shared__ double sr[256];
  double acc = 0.0;
  for (int i = threadIdx.x; i < nblocks; i += 256) acc += (double)partials[i];
  sr[threadIdx.x] = acc;
  __syncthreads();
  #pragma unroll
  for (int s = 128; s > 0; s >>= 1) {
    if (threadIdx.x < s) sr[threadIdx.x] += sr[threadIdx.x + s];
    __syncthreads();
  }
  if (threadIdx.x == 0) {
    double mean = sr[0] / (double)(64.0 * 512.0 * 512.0);
    out[0] = (float)(1.0 - mean);
  }
}

extern "C" void kernel_launch(void* const* d_in, const int* in_sizes, int n_in,
                              void* d_out, int out_size, void* d_ws,
                              size_t ws_size, hipStream_t stream) {
  (void)in_sizes; (void)n_in; (void)out_size; (void)ws_size;
  const float* pred   = (const float*)d_in[0];
  const float* target = (const float*)d_in[1];
  float* out      = (float*)d_out;
  float* partials = (float*)d_ws;   // 4096 floats, fully overwritten each call

  const int nblocks = 64 * 8 * 8;   // B * (512/64)^2
  ssim_main_kernel<<<nblocks, 256, 0, stream>>>(pred, target, partials);
  ssim_finalize_kernel<<<1, 256, 0, stream>>>(partials, out, nblocks);
}